// RNN_30056181137426
// MI455X (gfx1250) — compile-verified
//
#include <hip/hip_runtime.h>
#include <math.h>

#define H 4096
#define IN 512
#define OUT 64
#define KSLICES 16
#define KCHUNK (H / KSLICES) // 256 rows per K-slice

typedef __attribute__((ext_vector_type(2))) float v2f;
typedef __attribute__((ext_vector_type(4))) float v4f;
typedef __attribute__((ext_vector_type(8))) float v8f;

// ---------------------------------------------------------------------------
// Kernel A: base[j] = b_i2h[j] + dot(inp, W_i2h[j,:])   (one wave per row j)
// ---------------------------------------------------------------------------
__global__ __launch_bounds__(256) void k_base(const float* __restrict__ inp,
                                              const float* __restrict__ W_i2h,
                                              const float* __restrict__ b_i2h,
                                              float* __restrict__ base) {
  const int wave = threadIdx.x >> 5;
  const int lane = threadIdx.x & 31;
  const int j = blockIdx.x * 8 + wave;
  const float* row = W_i2h + (size_t)j * IN;
  float s = 0.f;
#pragma unroll
  for (int k = lane; k < IN; k += 32) s += inp[k] * row[k];
#pragma unroll
  for (int off = 16; off > 0; off >>= 1) s += __shfl_down(s, off, 32);
  if (lane == 0) base[j] = s + b_i2h[j];
}

// ---------------------------------------------------------------------------
// Kernel B: partial[slice][j] = sum_{k in slice} hidden[k]*(w[k,j]+plas[k,j]*hebb[k,j])
// via V_WMMA_F32_16X16X4_F32. One wave per (16-column tile, K-slice).
// A = hidden[k..k+3] replicated across all 16 M rows:
//   A layout (16x4 f32): VGPR0 = {K0 lanes0-15, K2 lanes16-31}, VGPR1 = {K1, K3}
// B layout (4x16 f32):  VGPR0 = {K0 row lanes0-15, K2 row lanes16-31}, VGPR1 = {K1, K3}
// => every row of D holds the identical matvec partial for columns j0..j0+15.
// w/plas loaded non-temporal (single use); hebb loaded RT so it stays in the
// 192MB L2 for the hebb-update pass.
// ---------------------------------------------------------------------------
__global__ __launch_bounds__(128) void k_gemv_wmma(const float* __restrict__ w,
                                                   const float* __restrict__ plas,
                                                   const float* __restrict__ hebb,
                                                   const float* __restrict__ hidden,
                                                   float* __restrict__ partial) {
  const int wave  = threadIdx.x >> 5;
  const int lane  = threadIdx.x & 31;
  const int tile  = blockIdx.x * 4 + wave;   // 0..255 column tiles of 16
  const int slice = blockIdx.y;              // 0..KSLICES-1
  const int j0    = tile * 16;
  const int half  = lane >> 4;
  const int n     = lane & 15;
  const int k0    = slice * KCHUNK;

  v8f acc = {};
  for (int k = k0; k < k0 + KCHUNK; k += 4) {
    const int r0 = k + 2 * half;               // this lane-half's two K rows
    const size_t idx0 = (size_t)r0 * H + j0 + n;
    const size_t idx1 = idx0 + H;
    const float m0 = __builtin_nontemporal_load(w + idx0) +
                     __builtin_nontemporal_load(plas + idx0) * hebb[idx0];
    const float m1 = __builtin_nontemporal_load(w + idx1) +
                     __builtin_nontemporal_load(plas + idx1) * hebb[idx1];
    v2f B = {m0, m1};
    v2f A = {hidden[r0], hidden[r0 + 1]};
    acc = __builtin_amdgcn_wmma_f32_16x16x4_f32(
        /*neg_a=*/false, A, /*neg_b=*/false, B,
        /*c_mod=*/(short)0, acc, /*reuse_a=*/false, /*reuse_b=*/false);
  }
  // D VGPR0, lanes 0..15: M=0 row, N=lane  -> the 16 column partials
  if (lane < 16) partial[(size_t)slice * H + j0 + lane] = acc[0];
}

// ---------------------------------------------------------------------------
// Kernel C: x[j] = relu(base[j] + sum_s partial[s][j])   (fixed order: exact)
// ---------------------------------------------------------------------------
__global__ __launch_bounds__(256) void k_relu(const float* __restrict__ partial,
                                              float* __restrict__ x) {
  const int j = blockIdx.x * 256 + threadIdx.x;
  float s = x[j];
#pragma unroll
  for (int sl = 0; sl < KSLICES; ++sl) s += partial[(size_t)sl * H + j];
  x[j] = fmaxf(s, 0.f);
}

// ---------------------------------------------------------------------------
// Kernel D: hebb_new[i,j] = (1-learn)*hebb[i,j] + learn*hidden[i]*x[j]
// hebb read should hit L2 (kept resident by NT-tagging w/plas in pass 1);
// hebb_new written non-temporal (never re-read).
// ---------------------------------------------------------------------------
__global__ __launch_bounds__(256) void k_hebb(const float* __restrict__ hebb,
                                              const float* __restrict__ hidden,
                                              const float* __restrict__ x,
                                              const float* __restrict__ learn,
                                              float* __restrict__ hebb_new) {
  const int i  = blockIdx.y;
  const int j4 = (blockIdx.x * 256 + threadIdx.x) * 4;
  const float lr   = learn[0];
  const float keep = 1.f - lr;
  const float hv   = hidden[i] * lr;
  const size_t idx = (size_t)i * H + j4;
  v4f hb = *(const v4f*)(hebb + idx);
  v4f xv = *(const v4f*)(x + j4);
  v4f r  = hb * keep + xv * hv;
  __builtin_nontemporal_store(r, (v4f*)(hebb_new + idx));
}

// ---------------------------------------------------------------------------
// Kernel E: out[o] = tanh(dot(x, W_h2o[o,:])), b_h2o unused   (block per output)
// ---------------------------------------------------------------------------
__global__ __launch_bounds__(256) void k_out(const float* __restrict__ x,
                                             const float* __restrict__ W_h2o,
                                             const float* __restrict__ b_h2o,
                                             float* __restrict__ out) {
  __shared__ float red[256];
  const int o = blockIdx.x;
  const float* row = W_h2o + (size_t)o * H;
  float s = 0.f;
  for (int j = threadIdx.x; j < H; j += 256) s += x[j] * row[j];
  red[threadIdx.x] = s;
  __syncthreads();
#pragma unroll
  for (int off = 128; off > 0; off >>= 1) {
    if (threadIdx.x < off) red[threadIdx.x] += red[threadIdx.x + off];
    __syncthreads();
  }
  if (threadIdx.x == 0) out[o] = tanhf(red[0]);
}

// ---------------------------------------------------------------------------
extern "C" void kernel_launch(void* const* d_in, const int* in_sizes, int n_in,
                              void* d_out, int out_size, void* d_ws, size_t ws_size,
                              hipStream_t stream) {
  (void)in_sizes; (void)n_in; (void)out_size; (void)d_ws; (void)ws_size;

  const float* inp    = (const float*)d_in[0];
  const float* hidden = (const float*)d_in[1];
  const float* hebb   = (const float*)d_in[2];
  const float* W_i2h  = (const float*)d_in[3];
  const float* b_i2h  = (const float*)d_in[4];
  const float* w      = (const float*)d_in[5];
  const float* plas   = (const float*)d_in[6];
  const float* learn  = (const float*)d_in[7];
  const float* W_h2o  = (const float*)d_in[8];
  const float* b_h2o  = (const float*)d_in[9];

  // d_out layout: [out(64) | x(4096) | hebb_new(4096*4096)]
  float* out_p    = (float*)d_out;
  float* x_p      = out_p + OUT;
  float* hebbn_p  = x_p + H;
  // Split-K partials live at the front of the hebb_new region; they are
  // consumed by k_relu and then fully overwritten by k_hebb. Deterministic:
  // written then read in stream order every call, no atomics anywhere.
  float* partial  = hebbn_p;

  k_base<<<H / 8, 256, 0, stream>>>(inp, W_i2h, b_i2h, x_p);
  k_gemv_wmma<<<dim3(H / 16 / 4, KSLICES), 128, 0, stream>>>(w, plas, hebb, hidden, partial);
  k_relu<<<H / 256, 256, 0, stream>>>(partial, x_p);
  k_hebb<<<dim3(H / 1024, H), 256, 0, stream>>>(hebb, hidden, x_p, learn, hebbn_p);
  k_out<<<OUT, 256, 0, stream>>>(x_p, W_h2o, b_h2o, out_p);
}